// ModelNew_69879117906365
// MI455X (gfx1250) — compile-verified
//
#include <hip/hip_runtime.h>

// Problem constants (from reference): B=2, T=2048, C=768, H=12, D=64
constexpr int Bn  = 2;
constexpr int Tn  = 2048;
constexpr int Cn  = 768;
constexpr int Hn  = 12;
constexpr int Dn  = 64;
constexpr int C3  = 3 * Cn;          // 2304
constexpr int Mtot = Bn * Tn;        // 4096 rows of qkv GEMM

typedef float v2f __attribute__((ext_vector_type(2)));
typedef float v8f __attribute__((ext_vector_type(8)));
typedef unsigned int u32;
typedef u32 v4u __attribute__((ext_vector_type(4)));
typedef int v4i __attribute__((ext_vector_type(4)));
typedef int v8i __attribute__((ext_vector_type(8)));

// fp32 WMMA: D(16x16,f32) = A(16x4,f32) * B(4x16,f32) + C
// A layout (per ISA): lane<16 -> row=lane,   holds K = k0+0, k0+1
//                     lane>=16 -> row=lane-16, holds K = k0+2, k0+3
// B layout (mirror):  lane<16 -> col=lane,   holds K = k0+0 (v0), k0+1 (v1)
//                     lane>=16 -> col=lane-16, holds K = k0+2 (v0), k0+3 (v1)
// C/D layout: vgpr g, lane<16 -> (m=g,   n=lane); lane>=16 -> (m=g+8, n=lane-16)
__device__ __forceinline__ v8f wmma_f32(v2f a, v2f b, v8f c) {
    return __builtin_amdgcn_wmma_f32_16x16x4_f32(
        /*neg_a=*/false, a, /*neg_b=*/false, b,
        /*c_mod=*/(short)0, c, /*reuse_a=*/false, /*reuse_b=*/false);
}

// ---------------------------------------------------------------------------
// Tensor Data Mover: DMA one 16-row x 64-col fp32 tile (4 KB, contiguous
// rows of 64 floats) from global memory into LDS at lds_byte_off.
// Descriptor packed per CDNA5 ISA (2-D tensor,
// groups 2/3 zero).  Tracked with TENSORcnt.
// Toolchain note: 6-arg builtin form (uint32x4, int32x8, int32x4, int32x4,
// int32x8, i32 cpol) — extra group zero-filled.
// ---------------------------------------------------------------------------
__device__ __forceinline__ void tdm_load_16x64_f32(const float* gsrc,
                                                   u32 lds_byte_off) {
    unsigned long long ga = (unsigned long long)(const void*)gsrc;
    v4u g0;
    g0[0] = 1u;                                          // count=1, user mode
    g0[1] = lds_byte_off;                                // lds_addr
    g0[2] = (u32)ga;                                     // global_addr[31:0]
    g0[3] = ((u32)(ga >> 32) & 0x01FFFFFFu) | (2u << 30); // addr[56:32] | type=2
    v8i g1;
    g1[0] = 0x20000;        // workgroup_mask=0, data_size=2 (4 bytes)
    g1[1] = 64 << 16;       // tensor_dim0 = 64    (bits 79:48, low half)
    g1[2] = 16 << 16;       // tensor_dim1 = 16    (bits 111:80, low half)
    g1[3] = 64 << 16;       // tile_dim0   = 64    (bits 127:112)
    g1[4] = 16;             // tile_dim1=16, tile_dim2=0
    g1[5] = 64;             // tensor_dim0_stride = 64 (bits 207:160, low)
    g1[6] = 0;              // dim0_stride hi, dim1_stride lo
    g1[7] = 0;              // dim1_stride hi (unused for 2-D)
    v4i z4 = {0, 0, 0, 0};                   // groups 2/3: 2-D tensor
    v8i z8 = {0, 0, 0, 0, 0, 0, 0, 0};       // extra group (unused, zero)
    __builtin_amdgcn_tensor_load_to_lds(g0, g1, z4, z4, z8, 0);
}

// ---------------------------------------------------------------------------
// Kernel 1: qkv = x @ W^T + b, scattered into q/k/v with [B,H,T,D] layout.
// One wave computes a 32(M) x 64(N) tile; each W fragment feeds 2 M-tiles,
// each x fragment feeds 4 N-tiles: 6 b64 loads per 8 WMMAs.
// Grid: (Mtot/32) * (C3/64) = 128 * 36 = 4608 wave-blocks.
// ---------------------------------------------------------------------------
__global__ __launch_bounds__(32) void qkv_gemm_kernel(
    const float* __restrict__ x,     // [B*T, C]
    const float* __restrict__ W,     // [3C, C] row-major
    const float* __restrict__ bias,  // [3C]
    float* __restrict__ qout,        // [B,H,T,D]
    float* __restrict__ kout,        // [B,H,T,D]
    float* __restrict__ vout)        // [B,H,T,D]
{
    const int lane = threadIdx.x;
    const int half = lane >> 4;
    const int l16  = lane & 15;

    const int ntilesN = C3 / 64;               // 36
    const int mtile = blockIdx.x / ntilesN;    // 0..127
    const int ntile = blockIdx.x % ntilesN;
    const int m0 = mtile * 32;
    const int n0 = ntile * 64;

    v8f acc[2][4] = {};

    const float* xrow0 = x + (size_t)(m0 +  0 + l16) * Cn;
    const float* xrow1 = x + (size_t)(m0 + 16 + l16) * Cn;
    const float* wrow0 = W + (size_t)(n0 +  0 + l16) * Cn;
    const float* wrow1 = W + (size_t)(n0 + 16 + l16) * Cn;
    const float* wrow2 = W + (size_t)(n0 + 32 + l16) * Cn;
    const float* wrow3 = W + (size_t)(n0 + 48 + l16) * Cn;

    for (int kk = 0; kk < Cn / 4; ++kk) {
        const int kb = kk * 4 + 2 * half;
        v2f a0 = *(const v2f*)(xrow0 + kb);
        v2f a1 = *(const v2f*)(xrow1 + kb);
        v2f b0 = *(const v2f*)(wrow0 + kb);
        v2f b1 = *(const v2f*)(wrow1 + kb);
        v2f b2 = *(const v2f*)(wrow2 + kb);
        v2f b3 = *(const v2f*)(wrow3 + kb);
        acc[0][0] = wmma_f32(a0, b0, acc[0][0]);
        acc[0][1] = wmma_f32(a0, b1, acc[0][1]);
        acc[0][2] = wmma_f32(a0, b2, acc[0][2]);
        acc[0][3] = wmma_f32(a0, b3, acc[0][3]);
        acc[1][0] = wmma_f32(a1, b0, acc[1][0]);
        acc[1][1] = wmma_f32(a1, b1, acc[1][1]);
        acc[1][2] = wmma_f32(a1, b2, acc[1][2]);
        acc[1][3] = wmma_f32(a1, b3, acc[1][3]);
    }

    // A 64-wide N tile lies in exactly one of {q,k,v} and one head.
    const int nsec = n0 / Cn;                  // 0=q, 1=k, 2=v
    const int c0   = n0 % Cn;
    const int h    = c0 / Dn;
    float* dst = (nsec == 0) ? qout : (nsec == 1) ? kout : vout;

#pragma unroll
    for (int mt = 0; mt < 2; ++mt) {
#pragma unroll
        for (int nt = 0; nt < 4; ++nt) {
            const int d = (c0 % Dn) + nt * 16 + l16;
            const float bv = bias[n0 + nt * 16 + l16];
#pragma unroll
            for (int g = 0; g < 8; ++g) {
                const int m  = m0 + mt * 16 + g + half * 8;  // row (b*T + t)
                const int bb = m / Tn;
                const int t  = m % Tn;
                dst[(((size_t)bb * Hn + h) * Tn + t) * Dn + d] =
                    acc[mt][nt][g] + bv;
            }
        }
    }
}

// ---------------------------------------------------------------------------
// Kernel 2: per (b,h), per 16-query tile:
//   K/V 16x64 tiles staged into LDS via TENSOR_LOAD_TO_LDS (double-buffered,
//   synced with s_wait_tensorcnt),
//   S = Q * K^T (WMMA over D=64), P = relu(tril(S))*scale (in regs),
//   P transposed D-layout -> A-layout via padded LDS,
//   Y += P * V (WMMA over 16-key contraction), streamed over key blocks <= qt.
// Grid: B*H*(T/16) = 3072 wave-blocks.
// ---------------------------------------------------------------------------

// Static LDS layout (floats). Compile-only assumption: static __shared__
// block is placed at LDS offset 0 (standard HIP placement).
//   kbuf[b] : floats [b*1024 .. )        bytes b*4096
//   vbuf[b] : floats [2048 + b*1024 .. ) bytes 8192 + b*4096
//   pbuf    : floats [4096 .. 4096+288)  bytes 16384
__global__ __launch_bounds__(32) void relu_attn_kernel(
    const float* __restrict__ q,   // [B,H,T,D]
    const float* __restrict__ k,   // [B,H,T,D]
    const float* __restrict__ v,   // [B,H,T,D]
    float* __restrict__ out)       // [B,T,C]
{
    __shared__ __align__(16) float smem[2 * 1024 + 2 * 1024 + 16 * 18];

    const int lane = threadIdx.x;
    const int half = lane >> 4;
    const int l16  = lane & 15;

    const int qtiles = Tn / 16;           // 128
    const int qt = blockIdx.x % qtiles;
    const int bh = blockIdx.x / qtiles;
    const int q0 = qt * 16;
    const float scale = 0.125f;           // 1/sqrt(64)

    const float* Q = q + (size_t)bh * Tn * Dn;
    const float* K = k + (size_t)bh * Tn * Dn;
    const float* V = v + (size_t)bh * Tn * Dn;

    // Preload Q tile as 16 A-fragments (D=64 contraction, 4 per step)
    v2f aq[16];
    const float* qrow = Q + (size_t)(q0 + l16) * Dn + 2 * half;
#pragma unroll
    for (int kk = 0; kk < 16; ++kk)
        aq[kk] = *(const v2f*)(qrow + kk * 4);

    v8f accy[4] = {};

    // Prologue: DMA the first K/V tiles into buffer 0.
    tdm_load_16x64_f32(K, /*kbuf0*/ 0u);
    tdm_load_16x64_f32(V, /*vbuf0*/ 8192u);

    for (int j = 0; j <= qt; ++j) {
        const int k0 = j * 16;
        const int buf = j & 1;

        if (j < qt) {
            // Buffer (j+1)&1 was consumed in iteration j-1; make sure all its
            // DS reads have retired before the TDM overwrites it.
            asm volatile("s_wait_dscnt 0x0" ::: "memory");
            const int nb = (j + 1) & 1;
            tdm_load_16x64_f32(K + (size_t)(k0 + 16) * Dn, (u32)(nb * 4096));
            tdm_load_16x64_f32(V + (size_t)(k0 + 16) * Dn, (u32)(8192 + nb * 4096));
            // Two TDM ops just issued for j+1; current tile ready when <= 2.
            __builtin_amdgcn_s_wait_tensorcnt((short)2);
        } else {
            __builtin_amdgcn_s_wait_tensorcnt((short)0);
        }

        const float* kb = &smem[buf * 1024];          // K tile [16][64]
        const float* vb = &smem[2048 + buf * 1024];   // V tile [16][64]
        float*       pb = &smem[4096];                // P transpose buffer

        // ---- S = Q * K^T : B-frag lane holds key column (l16), elems d
        v8f s = {};
        const float* krow = kb + l16 * 64 + 2 * half;
#pragma unroll
        for (int kk = 0; kk < 16; ++kk) {
            v2f bkf = *(const v2f*)(krow + kk * 4);
            s = wmma_f32(aq[kk], bkf, s);
        }

        // ---- mask (causal) + scale + relu, in C/D layout
#pragma unroll
        for (int g = 0; g < 8; ++g) {
            const int qi = q0 + g + 8 * half;  // query row
            const int ki = k0 + l16;           // key col
            float sv = s[g] * scale;
            sv = (qi >= ki && sv > 0.0f) ? sv : 0.0f;
            s[g] = sv;
        }

        // ---- D-layout -> A-layout transpose through LDS (padded stride 18)
        __syncthreads();
#pragma unroll
        for (int g = 0; g < 8; ++g)
            pb[(g + 8 * half) * 18 + l16] = s[g];
        __syncthreads();

        v2f ap[4];
#pragma unroll
        for (int kk = 0; kk < 4; ++kk)
            ap[kk] = *(const v2f*)(&pb[l16 * 18 + kk * 4 + 2 * half]);

        // ---- Y += P * V : contraction over 16 keys, 4 N-tiles of 16 dims
#pragma unroll
        for (int nt = 0; nt < 4; ++nt) {
#pragma unroll
            for (int kk = 0; kk < 4; ++kk) {
                const float* vp = vb + (kk * 4 + 2 * half) * 64 + nt * 16 + l16;
                v2f bvf;
                bvf.x = vp[0];     // key row r
                bvf.y = vp[64];    // key row r+1
                accy[nt] = wmma_f32(ap[kk], bvf, accy[nt]);
            }
        }
    }

    // ---- write Y back to [B,T,C]
    const int bb = bh / Hn;
    const int h  = bh % Hn;
#pragma unroll
    for (int nt = 0; nt < 4; ++nt) {
        const int col = h * Dn + nt * 16 + l16;
#pragma unroll
        for (int g = 0; g < 8; ++g) {
            const int t = q0 + g + 8 * half;
            out[((size_t)bb * Tn + t) * Cn + col] = accy[nt][g];
        }
    }
}

// ---------------------------------------------------------------------------
extern "C" void kernel_launch(void* const* d_in, const int* in_sizes, int n_in,
                              void* d_out, int out_size, void* d_ws, size_t ws_size,
                              hipStream_t stream) {
    (void)in_sizes; (void)n_in; (void)out_size; (void)ws_size;

    const float* x    = (const float*)d_in[0];   // [B,T,C]
    const float* W    = (const float*)d_in[1];   // [3C,C]
    const float* bias = (const float*)d_in[2];   // [3C]

    float* ws = (float*)d_ws;
    const size_t sec = (size_t)Bn * Hn * Tn * Dn;   // elements per q/k/v
    float* qb = ws;
    float* kb = ws + sec;
    float* vb = ws + 2 * sec;

    dim3 blk(32);
    const int gemm_blocks = (Mtot / 32) * (C3 / 64);      // 4608
    qkv_gemm_kernel<<<gemm_blocks, blk, 0, stream>>>(x, W, bias, qb, kb, vb);

    const int attn_blocks = Bn * Hn * (Tn / 16);          // 3072
    relu_attn_kernel<<<attn_blocks, blk, 0, stream>>>(qb, kb, vb, (float*)d_out);
}